// ParallelRetention_reduce_46643344835367
// MI455X (gfx1250) — compile-verified
//
#include <hip/hip_runtime.h>
#include <hip/hip_bf16.h>

typedef float v2f __attribute__((ext_vector_type(2)));
typedef float v8f __attribute__((ext_vector_type(8)));

#define NH    8
#define SEQ   4096
#define DIM   64
#define CHUNK 64
#define NC    (SEQ / CHUNK)
#define NT    256   // 8 waves of 32

// ---------------------------------------------------------------------------
// Pass 1: per-chunk summaries.  S_kv[h][c] = k_bar^T @ v (64x64),
//         ksum[h][c][d] = sum_r k_bar[r][d]
// ---------------------------------------------------------------------------
__global__ __launch_bounds__(NT) void pr_pass1(
    const float* __restrict__ kmat, const float* __restrict__ vmat,
    const float* __restrict__ coef_k,
    float* __restrict__ chunkKV, float* __restrict__ chunkKs)
{
  __shared__ float kb[CHUNK][DIM];   // k_bar (row-major); WMMA reads are row accesses
  __shared__ float vv[CHUNK][DIM];

  const int blk = blockIdx.x;
  const int h = blk / NC, c = blk % NC;
  const int s0 = c * CHUNK;
  const int tid = threadIdx.x;

  for (int e = tid; e < CHUNK * DIM; e += NT) {
    const int r = e >> 6, d = e & 63;
    const float ck = coef_k[h * SEQ + s0 + r];
    const size_t g = ((size_t)(h * SEQ + s0 + r)) * DIM + d;
    kb[r][d] = kmat[g] * ck;
    vv[r][d] = vmat[g];
  }
  __syncthreads();

  if (tid < DIM) {
    float a = 0.f;
    for (int r = 0; r < CHUNK; ++r) a += kb[r][tid];
    chunkKs[(h * NC + c) * DIM + tid] = a;
  }

  const int wave = tid >> 5, lane = tid & 31;
  const int lh = lane >> 4, lm = lane & 15;
  float* dst = chunkKV + ((size_t)(h * NC + c)) * DIM * DIM;

  // S_kv = kb^T @ vv : A[m][k] = kb[k][m], B[k][n] = vv[k][n]
  for (int t = wave * 2; t < wave * 2 + 2; ++t) {
    const int tm = (t >> 2) * 16, tn = (t & 3) * 16;
    v8f acc = {};
    for (int kk = 0; kk < CHUNK; kk += 4) {
      const int kv = kk + lh * 2;
      v2f a, b;
      a.x = kb[kv + 0][tm + lm];
      a.y = kb[kv + 1][tm + lm];
      b.x = vv[kv + 0][tn + lm];
      b.y = vv[kv + 1][tn + lm];
      acc = __builtin_amdgcn_wmma_f32_16x16x4_f32(false, a, false, b,
                                                  (short)0, acc, false, false);
    }
    for (int r = 0; r < 8; ++r)
      dst[(tm + r + 8 * lh) * DIM + tn + lm] = acc[r];
  }
}

// ---------------------------------------------------------------------------
// Pass 2: exclusive prefix over chunks (per head).  State is 64x64 + 64.
// ---------------------------------------------------------------------------
__global__ __launch_bounds__(NT) void pr_pass2(
    const float* __restrict__ chunkKV, const float* __restrict__ chunkKs,
    float* __restrict__ preKV, float* __restrict__ preKs)
{
  const int h = blockIdx.x;
  const int tid = threadIdx.x;

  float acc[16];
  for (int i = 0; i < 16; ++i) acc[i] = 0.f;

  for (int c = 0; c < NC; ++c) {
    const size_t base = ((size_t)(h * NC + c)) * DIM * DIM;
    for (int i = 0; i < 16; ++i) {
      const int e = tid + i * NT;
      preKV[base + e] = acc[i];
      acc[i] += chunkKV[base + e];
    }
  }

  if (tid < DIM) {
    float a = 0.f;
    for (int c = 0; c < NC; ++c) {
      preKs[(h * NC + c) * DIM + tid] = a;
      a += chunkKs[(h * NC + c) * DIM + tid];
    }
  }
}

// ---------------------------------------------------------------------------
// Pass 3: scores + P + output + RMS norm.
// ---------------------------------------------------------------------------
__global__ __launch_bounds__(NT) void pr_pass3(
    const float* __restrict__ qmat, const float* __restrict__ kmat,
    const float* __restrict__ vmat,
    const float* __restrict__ coef_q, const float* __restrict__ coef_k,
    const float* __restrict__ normer,
    const float* __restrict__ preKV, const float* __restrict__ preKs,
    float* __restrict__ out)
{
  __shared__ float qb[CHUNK][DIM + 1];   // column-strided frag reads -> pad
  __shared__ float kb[CHUNK][DIM + 1];   // column-strided frag reads -> pad
  __shared__ float vv[CHUNK][DIM];       // row reads only
  __shared__ float st[DIM][DIM];         // row reads only
  __shared__ float sc[CHUNK][DIM + 1];   // scores, then reused for O
  __shared__ float kpre[DIM];
  __shared__ float rs[CHUNK];

  const int blk = blockIdx.x;
  const int h = blk / NC, c = blk % NC;
  const int s0 = c * CHUNK;
  const int tid = threadIdx.x;

  for (int e = tid; e < CHUNK * DIM; e += NT) {
    const int r = e >> 6, d = e & 63;
    const int srow = s0 + r;
    const float cq = coef_q[h * SEQ + srow] / normer[h * SEQ + srow];
    const float ck = coef_k[h * SEQ + srow];
    const size_t g = ((size_t)(h * SEQ + srow)) * DIM + d;
    qb[r][d] = qmat[g] * cq;
    kb[r][d] = kmat[g] * ck;
    vv[r][d] = vmat[g];
    st[r][d] = preKV[((size_t)(h * NC + c)) * DIM * DIM + e];
  }
  if (tid < DIM) kpre[tid] = preKs[(h * NC + c) * DIM + tid];
  __syncthreads();

  const int wave = tid >> 5, lane = tid & 31;
  const int lh = lane >> 4, lm = lane & 15;

  // scores = qb @ kb^T with causal mask:  A[m][k]=qb[m][k], B[k][n]=kb[n][k]
  for (int t = wave * 2; t < wave * 2 + 2; ++t) {
    const int tm = (t >> 2) * 16, tn = (t & 3) * 16;
    v8f acc = {};
    for (int kk = 0; kk < DIM; kk += 4) {
      const int kv = kk + lh * 2;
      v2f a, b;
      a.x = qb[tm + lm][kv + 0];
      a.y = qb[tm + lm][kv + 1];
      b.x = kb[tn + lm][kv + 0];
      b.y = kb[tn + lm][kv + 1];
      acc = __builtin_amdgcn_wmma_f32_16x16x4_f32(false, a, false, b,
                                                  (short)0, acc, false, false);
    }
    for (int r = 0; r < 8; ++r) {
      const int i = tm + r + 8 * lh, j = tn + lm;
      sc[i][j] = (j <= i) ? acc[r] : 0.f;
    }
  }
  __syncthreads();

  // O = sc_causal @ vv + qb @ st   (keep tiles in registers)
  v8f oacc[2];
  for (int ti = 0; ti < 2; ++ti) {
    const int t = wave * 2 + ti;
    const int tm = (t >> 2) * 16, tn = (t & 3) * 16;
    v8f acc = {};
    for (int kk = 0; kk < CHUNK; kk += 4) {
      const int kv = kk + lh * 2;
      v2f a, b;
      a.x = sc[tm + lm][kv + 0];
      a.y = sc[tm + lm][kv + 1];
      b.x = vv[kv + 0][tn + lm];
      b.y = vv[kv + 1][tn + lm];
      acc = __builtin_amdgcn_wmma_f32_16x16x4_f32(false, a, false, b,
                                                  (short)0, acc, false, false);
    }
    for (int kk = 0; kk < DIM; kk += 4) {
      const int kv = kk + lh * 2;
      v2f a, b;
      a.x = qb[tm + lm][kv + 0];
      a.y = qb[tm + lm][kv + 1];
      b.x = st[kv + 0][tn + lm];
      b.y = st[kv + 1][tn + lm];
      acc = __builtin_amdgcn_wmma_f32_16x16x4_f32(false, a, false, b,
                                                  (short)0, acc, false, false);
    }
    oacc[ti] = acc;
  }

  // P_i = row-sum of causal scores + q_bar_i . ksum_prefix ; clamp
  if (tid < CHUNK) {
    float p = 0.f;
    for (int j = 0; j < CHUNK; ++j) p += sc[tid][j];
    for (int d = 0; d < DIM; ++d) p += qb[tid][d] * kpre[d];
    p = fabsf(p);
    if (p < 1.f) p = 1.f;
    rs[tid] = 1.f / p;
  }
  __syncthreads();   // everyone done reading sc (and rs written)

  for (int ti = 0; ti < 2; ++ti) {
    const int t = wave * 2 + ti;
    const int tm = (t >> 2) * 16, tn = (t & 3) * 16;
    for (int r = 0; r < 8; ++r)
      sc[tm + r + 8 * lh][tn + lm] = oacc[ti][r];
  }
  __syncthreads();

  // fold 1/P into RMS scale: out = (O/P) * rsqrt(mean((O/P)^2) + 1e-6)
  if (tid < CHUNK) {
    const float invp = rs[tid];
    float ssq = 0.f;
    for (int d = 0; d < DIM; ++d) {
      const float o = sc[tid][d] * invp;
      ssq += o * o;
    }
    rs[tid] = invp * rsqrtf(ssq * (1.f / DIM) + 1e-6f);
  }
  __syncthreads();

  for (int e = tid; e < CHUNK * DIM; e += NT) {
    const int r = e >> 6, d = e & 63;
    out[((size_t)(h * SEQ + s0 + r)) * DIM + d] = sc[r][d] * rs[r];
  }
}

// ---------------------------------------------------------------------------
extern "C" void kernel_launch(void* const* d_in, const int* in_sizes, int n_in,
                              void* d_out, int out_size, void* d_ws, size_t ws_size,
                              hipStream_t stream) {
  const float* q  = (const float*)d_in[0];
  const float* k  = (const float*)d_in[1];
  const float* v  = (const float*)d_in[2];
  const float* cq = (const float*)d_in[3];
  const float* ck = (const float*)d_in[4];
  const float* mn = (const float*)d_in[5];
  float* out = (float*)d_out;

  float* ws      = (float*)d_ws;
  float* chunkKV = ws;                                        // NH*NC*DIM*DIM
  float* preKV   = chunkKV + (size_t)NH * NC * DIM * DIM;     // NH*NC*DIM*DIM
  float* chunkKs = preKV   + (size_t)NH * NC * DIM * DIM;     // NH*NC*DIM
  float* preKs   = chunkKs + (size_t)NH * NC * DIM;           // NH*NC*DIM

  pr_pass1<<<NH * NC, NT, 0, stream>>>(k, v, ck, chunkKV, chunkKs);
  pr_pass2<<<NH,      NT, 0, stream>>>(chunkKV, chunkKs, preKV, preKs);
  pr_pass3<<<NH * NC, NT, 0, stream>>>(q, k, v, cq, ck, mn, preKV, preKs, out);
}